// STCM_5935644803579
// MI455X (gfx1250) — compile-verified
//
#include <hip/hip_runtime.h>
#include <math.h>

#define Bq 2
#define Lq 5
#define Hq 100
#define Wq 352
#define Cq 256
#define Nq (Bq * Lq)              // 10 samples
#define INV_VOXDS (1.0f / 1.6f)   // 1/(VOXEL*DOWNSAMPLE)

typedef __attribute__((ext_vector_type(2))) float v2f;
typedef __attribute__((ext_vector_type(4))) float v4f;
typedef __attribute__((ext_vector_type(8))) float v8f;

// ---------------------------------------------------------------------------
// Kernel 1: theta[n] = inv(norm @ Hm @ norm_inv)[:2,:]  (closed form, affine)
// ---------------------------------------------------------------------------
__global__ void STCM_theta_kernel(const float* __restrict__ scm,
                                  float* __restrict__ theta) {
    int n = threadIdx.x;
    if (n >= Nq) return;
    const float* s = scm + n * 16;
    float r00 = s[0], r01 = s[1], m02 = s[3] * INV_VOXDS;
    float r10 = s[4], r11 = s[5], m12 = s[7] * INV_VOXDS;

    // center = [H/2, W/2]; t = center - R@center + M[:,2]
    const float cy = Hq * 0.5f, cx = Wq * 0.5f;
    float t0 = cy - (r00 * cy + r01 * cx) + m02;
    float t1 = cx - (r10 * cy + r11 * cx) + m12;

    const float a  = 2.0f / (Wq - 1.0f);   // norm[0][0]
    const float b  = 2.0f / (Hq - 1.0f);   // norm[1][1]
    // dst_norm = norm @ [[R,t],[0,0,1]] @ norm_inv  (affine):
    float p  = r00;
    float q  = (a / b) * r01;
    float u  = r00 + (a / b) * r01 + a * t0 - 1.0f;
    float rr = (b / a) * r10;
    float ss = r11;
    float v  = (b / a) * r10 + r11 + b * t1 - 1.0f;

    float inv = 1.0f / (p * ss - q * rr);
    float* th = theta + n * 6;
    th[0] =  ss * inv;
    th[1] = -q  * inv;
    th[2] = (q * v - ss * u) * inv;
    th[3] = -rr * inv;
    th[4] =  p  * inv;
    th[5] = (rr * u - p * v) * inv;
}

// ---------------------------------------------------------------------------
// Kernel 2: pe[m,j] = sum_k emb[idx[m],k] * lin_W[j,k] + lin_b[j]
// WMMA f32 16x16x4, one wave per 16-column tile, 64 K-steps.
// A fragment: lane holds row M=lane&15; VGPR0/1 = K = kbase, kbase+1,
//   kbase = (lane<16 ? 0 : 2)  -> contiguous float2 from A row.
// B fragment: lane holds col N=lane&15; same K split -> contiguous float2
//   from lin_W row (since B[k][j] = lin_W[j][k]).
// Padded rows 10..15: clamped gather; their accum rows are never stored and
// never mix with valid rows. EXEC stays all-ones around every WMMA.
// ---------------------------------------------------------------------------
__global__ void STCM_pe_kernel(const int* __restrict__ prior,
                               const float* __restrict__ emb,
                               const float* __restrict__ linW,
                               const float* __restrict__ linb,
                               float* __restrict__ pe) {
    int lane   = threadIdx.x;        // 0..31
    int half   = lane >> 4;          // 0 | 1
    int lane16 = lane & 15;
    int j0     = blockIdx.x << 4;    // column tile base
    int col    = j0 + lane16;

    int rowc = lane16 < (Nq - 1) ? lane16 : (Nq - 1);   // clamped A row
    int e    = prior[rowc] * 2;                          // RTE_RATIO
    const float* Arow = emb + (size_t)e * Cq;
    const float* Brow = linW + (size_t)col * Cq;
    int kbase = half * 2;

    v8f c = {0.f, 0.f, 0.f, 0.f, 0.f, 0.f, 0.f, 0.f};
    #pragma unroll 8
    for (int k0 = 0; k0 < Cq; k0 += 4) {
        v2f av = *(const v2f*)(Arow + k0 + kbase);
        v2f bv = *(const v2f*)(Brow + k0 + kbase);
        c = __builtin_amdgcn_wmma_f32_16x16x4_f32(
                /*neg_a=*/false, av, /*neg_b=*/false, bv,
                /*c_mod=*/(short)0, c, /*reuse_a=*/false, /*reuse_b=*/false);
    }

    float bias = linb[col];
    // D layout: VGPR r -> M = r + half*8, N = col
    #pragma unroll
    for (int r = 0; r < 8; ++r) {
        int m = r + half * 8;
        if (m < Nq) pe[(size_t)m * Cq + col] = c[r] + bias;
    }
}

// ---------------------------------------------------------------------------
// Kernel 3: fused (x + pe) -> bilinear warp -> out, all in [B,L,H,W,C] layout.
// 256 threads = 4 pixels x 64 threads; each thread: 4 channels (float4/b128).
// out = sum_i w_i*valid_i*x_i + pe * sum_i(w_i*valid_i)
// ---------------------------------------------------------------------------
__global__ __launch_bounds__(256) void STCM_warp_kernel(
        const float* __restrict__ x,
        const float* __restrict__ theta,
        const float* __restrict__ pe,
        float* __restrict__ out) {
    int t   = threadIdx.x;
    int pix = blockIdx.x * 4 + (t >> 6);
    int cg  = (t & 63) << 2;                 // channel base (4 per thread)

    int n   = pix / (Hq * Wq);
    int rem = pix - n * (Hq * Wq);
    int h   = rem / Wq;
    int w   = rem - h * Wq;

    const float* th = theta + n * 6;
    float gx = -1.0f + 2.0f * (float)w / (Wq - 1.0f);
    float gy = -1.0f + 2.0f * (float)h / (Hq - 1.0f);
    float sx = th[0] * gx + th[1] * gy + th[2];
    float sy = th[3] * gx + th[4] * gy + th[5];
    float ix = (sx + 1.0f) * 0.5f * (float)(Wq - 1);
    float iy = (sy + 1.0f) * 0.5f * (float)(Hq - 1);

    float x0f = floorf(ix), y0f = floorf(iy);
    float x1f = x0f + 1.0f, y1f = y0f + 1.0f;
    float wx1 = ix - x0f, wx0 = 1.0f - wx1;
    float wy1 = iy - y0f, wy0 = 1.0f - wy1;

    bool vx0 = (x0f >= 0.0f) && (x0f <= (float)(Wq - 1));
    bool vx1 = (x1f >= 0.0f) && (x1f <= (float)(Wq - 1));
    bool vy0 = (y0f >= 0.0f) && (y0f <= (float)(Hq - 1));
    bool vy1 = (y1f >= 0.0f) && (y1f <= (float)(Hq - 1));

    float w00 = (vx0 && vy0) ? wx0 * wy0 : 0.0f;
    float w10 = (vx1 && vy0) ? wx1 * wy0 : 0.0f;
    float w01 = (vx0 && vy1) ? wx0 * wy1 : 0.0f;
    float w11 = (vx1 && vy1) ? wx1 * wy1 : 0.0f;

    int xc0 = (int)fminf(fmaxf(x0f, 0.0f), (float)(Wq - 1));
    int xc1 = (int)fminf(fmaxf(x1f, 0.0f), (float)(Wq - 1));
    int yc0 = (int)fminf(fmaxf(y0f, 0.0f), (float)(Hq - 1));
    int yc1 = (int)fminf(fmaxf(y1f, 0.0f), (float)(Hq - 1));

    size_t nbase = (size_t)n * Hq * Wq;
    const v4f* p00 = (const v4f*)(x + ((nbase + (size_t)yc0 * Wq + xc0) * Cq + cg));
    const v4f* p10 = (const v4f*)(x + ((nbase + (size_t)yc0 * Wq + xc1) * Cq + cg));
    const v4f* p01 = (const v4f*)(x + ((nbase + (size_t)yc1 * Wq + xc0) * Cq + cg));
    const v4f* p11 = (const v4f*)(x + ((nbase + (size_t)yc1 * Wq + xc1) * Cq + cg));

    v4f v00 = *p00;
    v4f v10 = *p10;
    v4f v01 = *p01;
    v4f v11 = *p11;
    v4f pe4 = *(const v4f*)(pe + (size_t)n * Cq + cg);

    float wsum = w00 + w10 + w01 + w11;
    v4f acc = v00 * w00 + v10 * w10 + v01 * w01 + v11 * w11 + pe4 * wsum;

    v4f* po = (v4f*)(out + ((nbase + (size_t)h * Wq + w) * Cq + cg));
    __builtin_nontemporal_store(acc, po);
}

// ---------------------------------------------------------------------------
extern "C" void kernel_launch(void* const* d_in, const int* in_sizes, int n_in,
                              void* d_out, int out_size, void* d_ws, size_t ws_size,
                              hipStream_t stream) {
    const float* x     = (const float*)d_in[0];
    const int*   prior = (const int*)  d_in[1];
    // d_in[2] = mask (unused by reference)
    const float* scm   = (const float*)d_in[3];
    const float* linW  = (const float*)d_in[4];
    const float* linb  = (const float*)d_in[5];
    const float* emb   = (const float*)d_in[6];
    float* out = (float*)d_out;

    float* theta = (float*)d_ws;          // 10*6 floats
    float* pe    = (float*)d_ws + 256;    // 10*256 floats (1KB-offset, aligned)

    STCM_theta_kernel<<<1, 32, 0, stream>>>(scm, theta);
    STCM_pe_kernel<<<Cq / 16, 32, 0, stream>>>(prior, emb, linW, linb, pe);

    int npix = Nq * Hq * Wq;              // 352,000 (divisible by 4)
    STCM_warp_kernel<<<npix / 4, 256, 0, stream>>>(x, theta, pe, out);
}